// LatentODENetwork_77129022702252
// MI455X (gfx1250) — compile-verified
//
#include <hip/hip_runtime.h>

// ---------------------------------------------------------------------------
// Latent ODE network on MI455X (gfx1250, wave32).
// fp32 everywhere via V_WMMA_F32_16X16X4_F32: reference is fp32 and the
// adaptive dopri5 controller is error-sensitive, so matrix math stays in
// fp32 on the WMMA pipe. Workload is compute/L2 bound (~340 GFLOP vs ~270MB
// of compulsory HBM traffic ~= 12us at 23.3 TB/s); ODE weights live in LDS,
// encoder/decoder weights + activations stay L2-resident (192MB L2).
// tanh is branch-free (v_exp_f32 + v_rcp_f32) to keep EXEC full and the
// VALU stream co-executable with WMMA.
// ---------------------------------------------------------------------------

typedef __attribute__((ext_vector_type(2))) float v2f;
typedef __attribute__((ext_vector_type(8))) float v8f;

static constexpr int BATCH = 32768;
static constexpr int LAT   = 16;
static constexpr int HID   = 128;
static constexpr int NEL   = BATCH * LAT;

// Padded LDS strides (avoid 64-bank conflicts on 128/16-wide rows).
static constexpr int HS = HID + 4;   // 132
static constexpr int LS = LAT + 4;   // 20

// Dormand-Prince tableau
static constexpr float C_A21 = (float)(1.0/5.0);
static constexpr float C_A31 = (float)(3.0/40.0),  C_A32 = (float)(9.0/40.0);
static constexpr float C_A41 = (float)(44.0/45.0), C_A42 = (float)(-56.0/15.0), C_A43 = (float)(32.0/9.0);
static constexpr float C_A51 = (float)(19372.0/6561.0), C_A52 = (float)(-25360.0/2187.0),
                       C_A53 = (float)(64448.0/6561.0), C_A54 = (float)(-212.0/729.0);
static constexpr float C_A61 = (float)(9017.0/3168.0), C_A62 = (float)(-355.0/33.0),
                       C_A63 = (float)(46732.0/5247.0), C_A64 = (float)(49.0/176.0),
                       C_A65 = (float)(-5103.0/18656.0);
static constexpr float C_B1 = (float)(35.0/384.0), C_B3 = (float)(500.0/1113.0),
                       C_B4 = (float)(125.0/192.0), C_B5 = (float)(-2187.0/6784.0),
                       C_B6 = (float)(11.0/84.0);
static constexpr float C_E1 = (float)(35.0/384.0  - 5179.0/57600.0);
static constexpr float C_E3 = (float)(500.0/1113.0 - 7571.0/16695.0);
static constexpr float C_E4 = (float)(125.0/192.0  - 393.0/640.0);
static constexpr float C_E5 = (float)(-2187.0/6784.0 + 92097.0/339200.0);
static constexpr float C_E6 = (float)(11.0/84.0    - 187.0/2100.0);
static constexpr float C_E7 = (float)(-1.0/40.0);

__device__ __forceinline__ v8f wmma4(v2f a, v2f b, v8f c) {
  // D = A(16x4,f32) * B(4x16,f32) + C(16x16,f32)
  return __builtin_amdgcn_wmma_f32_16x16x4_f32(false, a, false, b, (short)0, c,
                                               false, false);
}

// Branch-free tanh: tanh(x) = 1 - 2/(exp(2x)+1).
// v_exp_f32 + v_rcp_f32; saturates to +-1 at +-inf, propagates NaN,
// ~1 ulp absolute error -- no EXEC divergence around the WMMA stream.
__device__ __forceinline__ float fast_tanh(float x) {
  const float TWO_LOG2E = 2.8853900817779268f;  // 2*log2(e)
  float e = __builtin_amdgcn_exp2f(x * TWO_LOG2E);
  return 1.0f - 2.0f * __builtin_amdgcn_rcpf(e + 1.0f);
}

// ---------------------------------------------------------------------------
// Generic GEMM: C[B x N] = act(A[B x K] @ W[K x N] + bias), row-major fp32.
// 8 waves/block, 1 wave per 16-row tile, TN 16-col tiles per wave.
// A layout per lane (16x4 fp32 A-matrix):  M = lane&15, K = (lane>>4)*2 + reg.
// B layout per lane (4x16 fp32 B-matrix):  N = lane&15, K = (lane>>4)*2 + reg.
// D layout: VGPR v holds rows (lane<16 ? v : v+8), col = lane&15.
// ---------------------------------------------------------------------------
template <int TN, int ACT>
__global__ __launch_bounds__(256) void gemm_wmma_kernel(
    const float* __restrict__ A, const float* __restrict__ W,
    const float* __restrict__ bias, float* __restrict__ C, int K, int N) {
  const int lane = threadIdx.x & 31;
  const int wave = threadIdx.x >> 5;
  const int row0 = (blockIdx.x * 8 + wave) * 16;
  const int n0   = blockIdx.y * 16 * TN;
  const int m    = lane & 15;
  const int half = lane >> 4;
  const int koff = half * 2;

  const v8f vzero = {0.f, 0.f, 0.f, 0.f, 0.f, 0.f, 0.f, 0.f};
  v8f acc[TN];
#pragma unroll
  for (int t = 0; t < TN; ++t) acc[t] = vzero;

  const float* Ar = A + (size_t)(row0 + m) * K;
  for (int k = 0; k < K; k += 4) {
    v2f a;
    a.x = Ar[k + koff];
    a.y = Ar[k + koff + 1];
#pragma unroll
    for (int t = 0; t < TN; ++t) {
      const float* Wp = W + (size_t)(k + koff) * N + (n0 + t * 16 + m);
      v2f b;
      b.x = Wp[0];
      b.y = Wp[N];
      acc[t] = wmma4(a, b, acc[t]);
    }
  }

  const int rbase = row0 + half * 8;
#pragma unroll
  for (int t = 0; t < TN; ++t) {
    const int n   = n0 + t * 16 + m;
    const float bv = bias[n];
#pragma unroll
    for (int v = 0; v < 8; ++v) {
      float val = acc[t][v] + bv;
      if (ACT == 1) val = fmaxf(val, 0.0f);
      C[(size_t)(rbase + v) * N + n] = val;
    }
  }
}

// ---------------------------------------------------------------------------
// Fused ODE vector field:  kout = W3^T tanh(W2^T tanh(W1^T zin + b1) + b2) + b3
// with zin = z + h * (c1*k1 + ... + c5*k5)  (null k pointers are skipped).
// If zin_out != null the combined input is also stored (used to emit y5 as a
// side product of the k7 stage, fusing the y5-combine kernel away).
// 8 waves/block, 16 rows/wave; weights staged in LDS once per block; per-wave
// LDS u-buffer re-layouts D tiles -> A tiles between layers.
// ---------------------------------------------------------------------------
__global__ __launch_bounds__(256) void ode_f_eval_kernel(
    const float* __restrict__ zb,
    const float* __restrict__ kp1, const float* __restrict__ kp2,
    const float* __restrict__ kp3, const float* __restrict__ kp4,
    const float* __restrict__ kp5,
    float c1, float c2, float c3, float c4, float c5,
    const float* __restrict__ scal,   // scal[2] = h
    const float* __restrict__ W1, const float* __restrict__ b1,
    const float* __restrict__ W2, const float* __restrict__ b2,
    const float* __restrict__ W3, const float* __restrict__ b3,
    float* __restrict__ kout, float* __restrict__ zin_out) {
  __shared__ float sW1[LAT * HS];
  __shared__ float sW2[HID * HS];
  __shared__ float sW3[HID * LS];
  __shared__ float sb1[HID];
  __shared__ float sb2[HID];
  __shared__ float sb3[LAT];
  __shared__ float sU[8][LAT * HS];

  const int tid = threadIdx.x;
  for (int i = tid; i < LAT * HID; i += 256) sW1[(i / HID) * HS + (i % HID)] = W1[i];
  for (int i = tid; i < HID * HID; i += 256) sW2[(i / HID) * HS + (i % HID)] = W2[i];
  for (int i = tid; i < HID * LAT; i += 256) sW3[(i / LAT) * LS + (i % LAT)] = W3[i];
  for (int i = tid; i < HID; i += 256) { sb1[i] = b1[i]; sb2[i] = b2[i]; }
  if (tid < LAT) sb3[tid] = b3[tid];
  __syncthreads();

  const int lane = tid & 31;
  const int wave = tid >> 5;
  const int row0 = (blockIdx.x * 8 + wave) * 16;
  const int m    = lane & 15;
  const int half = lane >> 4;
  const int koff = half * 2;
  const int hi8  = half * 8;
  const float h  = scal[2];
  float* uw = &sU[wave][0];
  const v8f vzero = {0.f, 0.f, 0.f, 0.f, 0.f, 0.f, 0.f, 0.f};

  // ---- build zin tile in A layout (K = 16 -> 4 chunks of 4) ----
  v2f za[4];
  const size_t rb = (size_t)(row0 + m) * LAT;
#pragma unroll
  for (int c = 0; c < 4; ++c) {
    const int kk = c * 4 + koff;
    float x0 = zb[rb + kk];
    float x1 = zb[rb + kk + 1];
    if (kp1) { x0 = fmaf(h * c1, kp1[rb + kk], x0); x1 = fmaf(h * c1, kp1[rb + kk + 1], x1); }
    if (kp2) { x0 = fmaf(h * c2, kp2[rb + kk], x0); x1 = fmaf(h * c2, kp2[rb + kk + 1], x1); }
    if (kp3) { x0 = fmaf(h * c3, kp3[rb + kk], x0); x1 = fmaf(h * c3, kp3[rb + kk + 1], x1); }
    if (kp4) { x0 = fmaf(h * c4, kp4[rb + kk], x0); x1 = fmaf(h * c4, kp4[rb + kk + 1], x1); }
    if (kp5) { x0 = fmaf(h * c5, kp5[rb + kk], x0); x1 = fmaf(h * c5, kp5[rb + kk + 1], x1); }
    za[c].x = x0;
    za[c].y = x1;
    if (zin_out) {
      zin_out[rb + kk]     = x0;
      zin_out[rb + kk + 1] = x1;
    }
  }

  // ---- layer 1: (16x16) @ (16x128), tanh -> u (LDS, row-major) ----
  for (int nt = 0; nt < 8; ++nt) {
    v8f acc = vzero;
#pragma unroll
    for (int c = 0; c < 4; ++c) {
      const int kk = c * 4 + koff;
      v2f b;
      b.x = sW1[kk * HS + nt * 16 + m];
      b.y = sW1[(kk + 1) * HS + nt * 16 + m];
      acc = wmma4(za[c], b, acc);
    }
    const int n   = nt * 16 + m;
    const float bv = sb1[n];
#pragma unroll
    for (int v = 0; v < 8; ++v)
      uw[(hi8 + v) * HS + n] = fast_tanh(acc[v] + bv);
  }

  // ---- preload u1 in A layout (K = 128 -> 32 chunks) so the LDS buffer
  //      can be overwritten in place by layer 2 output ----
  v2f ua[32];
#pragma unroll
  for (int c = 0; c < 32; ++c) {
    const int kk = c * 4 + koff;
    ua[c].x = uw[m * HS + kk];
    ua[c].y = uw[m * HS + kk + 1];
  }

  // ---- layer 2: (16x128) @ (128x128), tanh -> u (LDS) ----
  for (int nt = 0; nt < 8; ++nt) {
    v8f acc = vzero;
#pragma unroll
    for (int c = 0; c < 32; ++c) {
      const int kk = c * 4 + koff;
      v2f b;
      b.x = sW2[kk * HS + nt * 16 + m];
      b.y = sW2[(kk + 1) * HS + nt * 16 + m];
      acc = wmma4(ua[c], b, acc);
    }
    const int n   = nt * 16 + m;
    const float bv = sb2[n];
#pragma unroll
    for (int v = 0; v < 8; ++v)
      uw[(hi8 + v) * HS + n] = fast_tanh(acc[v] + bv);
  }

  // ---- layer 3: (16x128) @ (128x16) -> kout (global) ----
  v8f acc = vzero;
#pragma unroll
  for (int c = 0; c < 32; ++c) {
    const int kk = c * 4 + koff;
    v2f a;
    a.x = uw[m * HS + kk];
    a.y = uw[m * HS + kk + 1];
    v2f b;
    b.x = sW3[kk * LS + m];
    b.y = sW3[(kk + 1) * LS + m];
    acc = wmma4(a, b, acc);
  }
  const float bv = sb3[m];
#pragma unroll
  for (int v = 0; v < 8; ++v)
    kout[(size_t)(row0 + hi8 + v) * LAT + m] = acc[v] + bv;
}

// ---------------------------------------------------------------------------
// Dopri5 control / elementwise kernels
// ---------------------------------------------------------------------------
__global__ void ode_init_kernel(float* scal) {
  scal[0] = 0.0f;   // t
  scal[1] = 0.1f;   // dt (DT0)
  scal[2] = 0.0f;   // h
  scal[4] = 0.0f;   // update flag
}

__global__ void ctrl_begin_kernel(float* scal) {
  const float t = scal[0], dt = scal[1];
  scal[2] = fminf(dt, 1.0f - t);   // h = min(dt, t1 - t)
}

__global__ __launch_bounds__(256) void err_partial_kernel(
    const float* __restrict__ z, const float* __restrict__ y5,
    const float* __restrict__ k1, const float* __restrict__ k3,
    const float* __restrict__ k4, const float* __restrict__ k5,
    const float* __restrict__ k6, const float* __restrict__ k7,
    const float* __restrict__ scal, float* __restrict__ partials) {
  __shared__ float red[256];
  const float h = scal[2];
  float local = 0.f;
  for (int i = blockIdx.x * 256 + threadIdx.x; i < NEL; i += 256 * 256) {
    const float err = h * (C_E1 * k1[i] + C_E3 * k3[i] + C_E4 * k4[i] +
                           C_E5 * k5[i] + C_E6 * k6[i] + C_E7 * k7[i]);
    const float tol = 1e-4f + 1e-4f * fmaxf(fabsf(z[i]), fabsf(y5[i]));
    const float r = err / tol;
    local += r * r;
  }
  red[threadIdx.x] = local;
  __syncthreads();
  for (int s = 128; s > 0; s >>= 1) {
    if (threadIdx.x < s) red[threadIdx.x] += red[threadIdx.x + s];
    __syncthreads();
  }
  if (threadIdx.x == 0) partials[blockIdx.x] = red[0];
}

__global__ void ctrl_end_kernel(float* scal, const float* __restrict__ partials,
                                int nPart) {
  float s = 0.f;
  for (int i = 0; i < nPart; ++i) s += partials[i];  // fixed order: deterministic
  const float enorm = fmaxf(sqrtf(s / (float)NEL), 1e-10f);
  const float t = scal[0], dt = scal[1], h = scal[2];
  const bool accept = (enorm <= 1.0f);
  const bool done   = (t >= 1.0f - 1e-8f);
  const float factor = fminf(fmaxf(0.9f * powf(enorm, -0.2f), 0.2f), 10.0f);
  scal[0] = done ? t : (accept ? t + h : t);
  scal[1] = done ? dt : h * factor;
  scal[4] = (!done && accept) ? 1.0f : 0.0f;
}

__global__ void z_update_kernel(float* __restrict__ z,
                                const float* __restrict__ y5,
                                const float* __restrict__ scal) {
  const int i = blockIdx.x * 256 + threadIdx.x;
  if (i < NEL && scal[4] != 0.0f) z[i] = y5[i];
}

__global__ void copy_z_kernel(const float* __restrict__ z, float* __restrict__ out) {
  const int i = blockIdx.x * 256 + threadIdx.x;
  if (i < NEL) out[i] = z[i];
}

// ---------------------------------------------------------------------------
extern "C" void kernel_launch(void* const* d_in, const int* in_sizes, int n_in,
                              void* d_out, int out_size, void* d_ws, size_t ws_size,
                              hipStream_t stream) {
  (void)in_sizes; (void)n_in; (void)out_size; (void)ws_size;
  const float* x      = (const float*)d_in[0];
  const float* enc_w1 = (const float*)d_in[1];
  const float* enc_b1 = (const float*)d_in[2];
  const float* enc_w2 = (const float*)d_in[3];
  const float* enc_b2 = (const float*)d_in[4];
  const float* enc_w3 = (const float*)d_in[5];
  const float* enc_b3 = (const float*)d_in[6];
  const float* ode_w1 = (const float*)d_in[7];
  const float* ode_b1 = (const float*)d_in[8];
  const float* ode_w2 = (const float*)d_in[9];
  const float* ode_b2 = (const float*)d_in[10];
  const float* ode_w3 = (const float*)d_in[11];
  const float* ode_b3 = (const float*)d_in[12];
  const float* dec_w1 = (const float*)d_in[13];
  const float* dec_b1 = (const float*)d_in[14];
  const float* dec_w2 = (const float*)d_in[15];
  const float* dec_b2 = (const float*)d_in[16];
  const float* dec_w3 = (const float*)d_in[17];
  const float* dec_b3 = (const float*)d_in[18];

  // Workspace layout (bytes):
  //   [0, 32M)    bufA : enc/dec B x 256 buffer; reused for k1..k7,y5 in ODE
  //   [32M, 48M)  bufB : enc/dec B x 128 buffer
  //   [48M, 50M)  z    : B x 16 latent state
  //   [50M, ...)  scalars (t, dt, h, -, flag) + 256 reduction partials
  char* ws = (char*)d_ws;
  float* bufA = (float*)ws;
  float* bufB = (float*)(ws + (size_t)BATCH * 256 * 4);
  float* z    = (float*)(ws + (size_t)BATCH * 256 * 4 + (size_t)BATCH * 128 * 4);
  float* scal = (float*)((char*)z + (size_t)NEL * 4);
  float* partials = scal + 8;
  float* karr[7];
  for (int i = 0; i < 7; ++i) karr[i] = bufA + (size_t)i * NEL;
  float* y5 = bufA + (size_t)7 * NEL;

  const dim3 blk(256);
  const int rgrid = BATCH / 128;             // 256 blocks (8 row-tiles each)
  const int egrid = (NEL + 255) / 256;       // elementwise over B x L

  // ---- encoder ----
  gemm_wmma_kernel<4, 1><<<dim3(rgrid, 256 / 64), blk, 0, stream>>>(x, enc_w1, enc_b1, bufA, 1024, 256);
  gemm_wmma_kernel<4, 1><<<dim3(rgrid, 128 / 64), blk, 0, stream>>>(bufA, enc_w2, enc_b2, bufB, 256, 128);
  gemm_wmma_kernel<1, 0><<<dim3(rgrid, 1),        blk, 0, stream>>>(bufB, enc_w3, enc_b3, z, 128, 16);

  // ---- dopri5, 32 fixed iterations (reference scans MAX_STEPS) ----
  ode_init_kernel<<<1, 1, 0, stream>>>(scal);
  const float* nf = nullptr;
  float* nw = nullptr;
  for (int s = 0; s < 32; ++s) {
    ctrl_begin_kernel<<<1, 1, 0, stream>>>(scal);
    ode_f_eval_kernel<<<rgrid, blk, 0, stream>>>(z, nf, nf, nf, nf, nf,
        0.f, 0.f, 0.f, 0.f, 0.f, scal,
        ode_w1, ode_b1, ode_w2, ode_b2, ode_w3, ode_b3, karr[0], nw);
    ode_f_eval_kernel<<<rgrid, blk, 0, stream>>>(z, karr[0], nf, nf, nf, nf,
        C_A21, 0.f, 0.f, 0.f, 0.f, scal,
        ode_w1, ode_b1, ode_w2, ode_b2, ode_w3, ode_b3, karr[1], nw);
    ode_f_eval_kernel<<<rgrid, blk, 0, stream>>>(z, karr[0], karr[1], nf, nf, nf,
        C_A31, C_A32, 0.f, 0.f, 0.f, scal,
        ode_w1, ode_b1, ode_w2, ode_b2, ode_w3, ode_b3, karr[2], nw);
    ode_f_eval_kernel<<<rgrid, blk, 0, stream>>>(z, karr[0], karr[1], karr[2], nf, nf,
        C_A41, C_A42, C_A43, 0.f, 0.f, scal,
        ode_w1, ode_b1, ode_w2, ode_b2, ode_w3, ode_b3, karr[3], nw);
    ode_f_eval_kernel<<<rgrid, blk, 0, stream>>>(z, karr[0], karr[1], karr[2], karr[3], nf,
        C_A51, C_A52, C_A53, C_A54, 0.f, scal,
        ode_w1, ode_b1, ode_w2, ode_b2, ode_w3, ode_b3, karr[4], nw);
    ode_f_eval_kernel<<<rgrid, blk, 0, stream>>>(z, karr[0], karr[1], karr[2], karr[3], karr[4],
        C_A61, C_A62, C_A63, C_A64, C_A65, scal,
        ode_w1, ode_b1, ode_w2, ode_b2, ode_w3, ode_b3, karr[5], nw);
    // k7 stage: zin = z + h*(B1 k1 + B3 k3 + B4 k4 + B5 k5 + B6 k6) == y5,
    // stored as a side product (fuses the y5-combine kernel).
    ode_f_eval_kernel<<<rgrid, blk, 0, stream>>>(z, karr[0], karr[2], karr[3], karr[4], karr[5],
        C_B1, C_B3, C_B4, C_B5, C_B6, scal,
        ode_w1, ode_b1, ode_w2, ode_b2, ode_w3, ode_b3, karr[6], y5);
    err_partial_kernel<<<256, blk, 0, stream>>>(z, y5, karr[0], karr[2], karr[3],
                                                karr[4], karr[5], karr[6], scal, partials);
    ctrl_end_kernel<<<1, 1, 0, stream>>>(scal, partials, 256);
    z_update_kernel<<<egrid, blk, 0, stream>>>(z, y5, scal);
  }

  // ---- decoder (last layer writes x_hat straight into d_out) ----
  float* out = (float*)d_out;
  gemm_wmma_kernel<4, 1><<<dim3(rgrid, 128 / 64),  blk, 0, stream>>>(z, dec_w1, dec_b1, bufB, 16, 128);
  gemm_wmma_kernel<4, 1><<<dim3(rgrid, 256 / 64),  blk, 0, stream>>>(bufB, dec_w2, dec_b2, bufA, 128, 256);
  gemm_wmma_kernel<4, 0><<<dim3(rgrid, 1024 / 64), blk, 0, stream>>>(bufA, dec_w3, dec_b3, out, 256, 1024);

  // ---- z_T appended after x_hat ----
  copy_z_kernel<<<egrid, blk, 0, stream>>>(z, out + (size_t)BATCH * 1024);
}